// SimpleGCN_54125177864457
// MI455X (gfx1250) — compile-verified
//
#include <hip/hip_runtime.h>

// ---------------------------------------------------------------------------
// GCN (2-layer, GCNConv w/ self loops + mean pool + linear head) for gfx1250.
// Memory-bound on edge gather/scatter (~1.7 GB @ 23.3 TB/s); GEMMs use
// V_WMMA_F32_16X16X4_F32. Atomic scatter uses native global_atomic_add_f32.
// ---------------------------------------------------------------------------

#define NN 100000
#define EE 1600000
#define GG 512

typedef float v2f __attribute__((ext_vector_type(2)));
typedef float v8f __attribute__((ext_vector_type(8)));

__device__ __forceinline__ void atomAddF(float* p, float v) {
    // relaxed agent-scope fp32 add -> global_atomic_add_f32 (no CAS loop)
    __hip_atomic_fetch_add(p, v, __ATOMIC_RELAXED, __HIP_MEMORY_SCOPE_AGENT);
}

// --- init: deg = 1 (self loop weight), pooled = 0, counts = 0 --------------
__global__ __launch_bounds__(256) void gcn_init(float* deg, float* pooled, float* counts) {
    int i = blockIdx.x * 256 + threadIdx.x;
    if (i < NN) deg[i] = 1.0f;
    if (i < GG * 64) pooled[i] = 0.0f;
    if (i < GG) counts[i] = 0.0f;
}

// --- degree accumulation over edges (target = col = edge_index[1]) ---------
__global__ __launch_bounds__(256) void gcn_deg(const int* __restrict__ ei, float* deg) {
    int e = blockIdx.x * 256 + threadIdx.x;
    if (e < EE) atomAddF(&deg[ei[EE + e]], 1.0f);
}

// --- dinv = rsqrt(deg) (deg >= 1 always, matches where(deg>0)) -------------
__global__ __launch_bounds__(256) void gcn_rsqrt(const float* __restrict__ deg, float* dinv) {
    int i = blockIdx.x * 256 + threadIdx.x;
    if (i < NN) dinv[i] = rsqrtf(deg[i]);
}

// --- T = A[N,64] @ W[64,64] via V_WMMA_F32_16X16X4_F32 ---------------------
// block = 256 threads = 8 waves; block tile = 32 rows x 64 cols;
// wave (w) tile: rows (w>>2)*16, cols (w&3)*16; K in 16 steps of 4.
__global__ __launch_bounds__(256) void gcn_gemm64(const float* __restrict__ A,
                                                  const float* __restrict__ W,
                                                  float* __restrict__ T, int nrows) {
    __shared__ float Wl[64 * 64];   // 16 KB
    __shared__ float Al[32 * 64];   // 8 KB
    const int tid = threadIdx.x;
    for (int i = tid; i < 64 * 64; i += 256) Wl[i] = W[i];
    const int rowBase = blockIdx.x * 32;
    for (int i = tid; i < 32 * 64; i += 256) {
        int r = rowBase + (i >> 6);
        Al[i] = (r < nrows) ? A[r * 64 + (i & 63)] : 0.0f;
    }
    __syncthreads();

    const int wave  = tid >> 5;
    const int lane  = tid & 31;
    const int l     = lane & 15;
    const int half  = lane >> 4;       // 0: K={0,1}; 1: K={2,3}
    const int mT    = (wave >> 2) * 16;
    const int n0    = (wave & 3) * 16;

    v8f c = {};
    #pragma unroll
    for (int k = 0; k < 64; k += 4) {
        const int ka = k + 2 * half;
        v2f a, b;
        a.x = Al[(mT + l) * 64 + ka];          // A: row M=l, K=ka
        a.y = Al[(mT + l) * 64 + ka + 1];      //            K=ka+1
        b.x = Wl[ka * 64 + n0 + l];            // B: row K=ka,   N=n0+l
        b.y = Wl[(ka + 1) * 64 + n0 + l];      //    row K=ka+1
        c = __builtin_amdgcn_wmma_f32_16x16x4_f32(
                /*neg_a=*/false, a, /*neg_b=*/false, b,
                /*c_mod=*/(short)0, c, /*reuse_a=*/false, /*reuse_b=*/false);
    }

    const int gRow = rowBase + mT;
    #pragma unroll
    for (int v = 0; v < 8; ++v) {              // C VGPR v: row v + 8*half
        int r = gRow + v + 8 * half;
        if (r < nrows) T[r * 64 + n0 + l] = c[v];
    }
}

// --- agg[i] = T[i] * dinv[i]^2  (self-loop term seeds accumulator) ---------
__global__ __launch_bounds__(256) void gcn_selfloop(const float* __restrict__ T,
                                                    const float* __restrict__ dinv,
                                                    float* __restrict__ agg) {
    int i = blockIdx.x * 256 + threadIdx.x;    // i over N*64
    float di = dinv[i >> 6];
    agg[i] = T[i] * di * di;
}

// --- edge scatter: agg[col] += T[row] * dinv[row]*dinv[col] ----------------
// 64 threads (2 waves) per edge, one feature each; T-row reads coalesced.
__global__ __launch_bounds__(256) void gcn_scatter(const float* __restrict__ T,
                                                   const float* __restrict__ dinv,
                                                   const int* __restrict__ ei,
                                                   float* __restrict__ agg) {
    int t = blockIdx.x * 256 + threadIdx.x;
    int e = t >> 6;
    int d = t & 63;
    if (e >= EE) return;
    int r = ei[e];
    int c = ei[EE + e];
    float w = dinv[r] * dinv[c];
    atomAddF(&agg[c * 64 + d], T[r * 64 + d] * w);
}

// --- h = relu(agg + b), in place -------------------------------------------
__global__ __launch_bounds__(256) void gcn_bias_relu(float* __restrict__ agg,
                                                     const float* __restrict__ b) {
    int i = blockIdx.x * 256 + threadIdx.x;    // i over N*64
    float v = agg[i] + b[i & 63];
    agg[i] = v > 0.0f ? v : 0.0f;
}

// --- mean-pool accumulation: pooled[batch[i]] += h[i]; counts[g] += 1 ------
__global__ __launch_bounds__(256) void gcn_pool(const float* __restrict__ h,
                                                const int* __restrict__ batch,
                                                float* pooled, float* counts) {
    int i = blockIdx.x * 256 + threadIdx.x;    // i over N*64
    int node = i >> 6;
    int d = i & 63;
    int g = batch[node];
    atomAddF(&pooled[g * 64 + d], h[i]);
    if (d == 0) atomAddF(&counts[g], 1.0f);
}

// --- out[g] = (pooled[g]/max(count,1)) . Wfc + bfc -------------------------
__global__ __launch_bounds__(64) void gcn_final(const float* __restrict__ pooled,
                                                const float* __restrict__ counts,
                                                const float* __restrict__ Wfc,
                                                const float* __restrict__ bfc,
                                                float* __restrict__ out) {
    __shared__ float red[64];
    int g = blockIdx.x, d = threadIdx.x;
    red[d] = pooled[g * 64 + d] * Wfc[d];
    __syncthreads();
    for (int s = 32; s > 0; s >>= 1) {
        if (d < s) red[d] += red[d + s];
        __syncthreads();
    }
    if (d == 0) {
        float cnt = counts[g];
        if (cnt < 1.0f) cnt = 1.0f;
        out[g] = red[0] / cnt + bfc[0];
    }
}

extern "C" void kernel_launch(void* const* d_in, const int* in_sizes, int n_in,
                              void* d_out, int out_size, void* d_ws, size_t ws_size,
                              hipStream_t stream) {
    const float* x    = (const float*)d_in[0];
    const int*   ei   = (const int*)  d_in[1];  // [2, E]
    const int*   batch= (const int*)  d_in[2];
    const float* W1   = (const float*)d_in[3];
    const float* b1   = (const float*)d_in[4];
    const float* W2   = (const float*)d_in[5];
    const float* b2   = (const float*)d_in[6];
    const float* Wfc  = (const float*)d_in[7];
    const float* bfc  = (const float*)d_in[8];
    float*       out  = (float*)d_out;

    // workspace layout (floats): deg[N] dinv[N] t[N*64] agg[N*64] pooled[G*64] counts[G]
    float* ws     = (float*)d_ws;
    float* deg    = ws;
    float* dinv   = ws + NN;
    float* t      = ws + 2 * NN;
    float* agg    = ws + 2 * NN + 64 * NN;
    float* pooled = ws + 2 * NN + 128 * NN;
    float* counts = pooled + GG * 64;

    const int nd64  = (NN * 64) / 256;       // 25000
    const int nGemm = NN / 32;               // 3125
    const int nScat = (EE * 64) / 256;       // 400000

    gcn_init   <<<(NN + 255) / 256, 256, 0, stream>>>(deg, pooled, counts);
    gcn_deg    <<<EE / 256, 256, 0, stream>>>(ei, deg);
    gcn_rsqrt  <<<(NN + 255) / 256, 256, 0, stream>>>(deg, dinv);

    // layer 1
    gcn_gemm64   <<<nGemm, 256, 0, stream>>>(x, W1, t, NN);
    gcn_selfloop <<<nd64, 256, 0, stream>>>(t, dinv, agg);
    gcn_scatter  <<<nScat, 256, 0, stream>>>(t, dinv, ei, agg);
    gcn_bias_relu<<<nd64, 256, 0, stream>>>(agg, b1);

    // layer 2 (agg now holds h1; t reused)
    gcn_gemm64   <<<nGemm, 256, 0, stream>>>(agg, W2, t, NN);
    gcn_selfloop <<<nd64, 256, 0, stream>>>(t, dinv, agg);
    gcn_scatter  <<<nScat, 256, 0, stream>>>(t, dinv, ei, agg);
    gcn_bias_relu<<<nd64, 256, 0, stream>>>(agg, b2);

    // pooling + head
    gcn_pool <<<nd64, 256, 0, stream>>>(agg, batch, pooled, counts);
    gcn_final<<<GG, 64, 0, stream>>>(pooled, counts, Wfc, bfc, out);
}